// ExactModel_15092515078731
// MI455X (gfx1250) — compile-verified
//
#include <hip/hip_runtime.h>

// ---------------------------------------------------------------------------
// MI455X (gfx1250) implementation.
//   Stage 1: psi(t) = expm(-i t H) |0>  via Taylor series with the SPARSE H
//            (13 nonzeros/row), state resident in LDS. Replaces O(DIM^3) expm.
//   Stage 2: per (t, pauli-string): three 16x16 complex Kron-rotations applied
//            as (16x16)x(16x256) GEMMs on the WMMA pipe (v_wmma_f32_16x16x4_f32),
//            then |.|^2 and index gather. Everything LDS-resident.
// ---------------------------------------------------------------------------

typedef __attribute__((ext_vector_type(2))) float v2f;
typedef __attribute__((ext_vector_type(8))) float v8f;

#define NQ   12
#define DIM  4096
#define TKAY 96   // Taylor terms: t<1, ||H|| ~ O(20) -> converges to fp32 eps

__device__ __forceinline__ v8f wmma_f32_k4(v2f a, v2f b, v8f c) {
    // D(16x16,f32) = A(16x4,f32) * B(4x16,f32) + C
    return __builtin_amdgcn_wmma_f32_16x16x4_f32(
        /*neg_a=*/false, a, /*neg_b=*/false, b,
        /*c_mod=*/(short)0, c, /*reuse_a=*/false, /*reuse_b=*/false);
}

// Linear LDS index of element (row_along_contracted_axis, flattened_other_axes)
// for the state viewed as S[a][b][c] (16x16x16), linear = a*256 + b*16 + c.
// AX is a compile-time constant -> pure shift/add address math, no branches.
template <int AX>
__device__ __forceinline__ int sidx(int row, int col) {
    if constexpr (AX == 0) return (row << 8) + col;                        // contract a
    else if constexpr (AX == 1) return ((col >> 4) << 8) + (row << 4) + (col & 15); // contract b
    else return ((col >> 4) << 8) + ((col & 15) << 4) + row;               // contract c
}

// Entry (r,c) of Kron(u[p0],u[p1],u[p2],u[p3]); fully branchless (select-based).
__device__ __forceinline__ void kron_entry(const int* pq, int r, int c,
                                           float& mr, float& mi) {
    const float s = 0.70710678118654752440f;
    mr = 1.f; mi = 0.f;
#pragma unroll
    for (int j = 0; j < 4; j++) {
        const int   p  = pq[j];
        const float rb = (float)((r >> (3 - j)) & 1);
        const float cb = (float)((c >> (3 - j)) & 1);
        const float w0 = (p == 0) ? 1.f : 0.f;   // X basis [[s,s],[s,-s]]
        const float w1 = (p == 1) ? 1.f : 0.f;   // Y basis [[s,-is],[s,is]]
        const float w2 = 1.f - w0 - w1;          // Z basis: identity
        const float ur = w0 * s * (1.f - 2.f * rb * cb)
                       + w1 * s * (1.f - cb)
                       + w2 * (1.f - rb - cb + 2.f * rb * cb);
        const float ui = w1 * s * cb * (2.f * rb - 1.f);
        const float t1 = mr * ur - mi * ui;
        mi = mr * ui + mi * ur;
        mr = t1;
    }
}

// One axis application: dst = M_AX (16x16 complex) contracted with src along AX.
// Straight-line: 2 tiles/wave x (4+4+8) WMMAs, all uniform control flow.
template <int AX>
__device__ __forceinline__ void apply_axis(const float* __restrict__ srcR,
                                           const float* __restrict__ srcI,
                                           float* __restrict__ dstR,
                                           float* __restrict__ dstI,
                                           const int* __restrict__ pauli_row) {
    const int lane = threadIdx.x & 31;
    const int wave = threadIdx.x >> 5;   // 0..7
    const int lrow = lane & 15;
    const int lhi  = lane >> 4;          // 0 or 1

    int pq[4];
#pragma unroll
    for (int j = 0; j < 4; j++) pq[j] = pauli_row[AX * 4 + j];

    // A fragments per ISA 16x4 f32 layout: lane half selects K pair.
    v2f aR[4], aI[4];
#pragma unroll
    for (int kk = 0; kk < 4; kk++) {
        const int kb = kk * 4 + lhi * 2;
        float r0, i0, r1, i1;
        kron_entry(pq, lrow, kb,     r0, i0);
        kron_entry(pq, lrow, kb + 1, r1, i1);
        aR[kk].x = r0; aR[kk].y = r1;
        aI[kk].x = i0; aI[kk].y = i1;
    }

#pragma unroll
    for (int t2 = 0; t2 < 2; t2++) {
        const int tile = wave + t2 * 8;
        const int col  = (tile << 4) + lrow;     // B-fragment column (N=lane)
        v8f accR = {};   // Mr*Sr
        v8f accX = {};   // Mi*Si
        v8f accI = {};   // Mr*Si + Mi*Sr
#pragma unroll
        for (int kk = 0; kk < 4; kk++) {
            const int r0 = kk * 4 + lhi * 2;     // B rows for this lane-half
            const int j0 = sidx<AX>(r0,     col);
            const int j1 = sidx<AX>(r0 + 1, col);
            v2f bR, bI;
            bR.x = srcR[j0]; bR.y = srcR[j1];
            bI.x = srcI[j0]; bI.y = srcI[j1];
            accR = wmma_f32_k4(aR[kk], bR, accR);
            accX = wmma_f32_k4(aI[kk], bI, accX);
            accI = wmma_f32_k4(aR[kk], bI, accI);
            accI = wmma_f32_k4(aI[kk], bR, accI);
        }
        // Complex real part: Dr = Mr*Sr - Mi*Si
#pragma unroll
        for (int v = 0; v < 8; v++) accR[v] -= accX[v];
        // C/D layout: VGPR v -> row v (lanes 0-15) / v+8 (lanes 16-31)
#pragma unroll
        for (int v = 0; v < 8; v++) {
            const int m = v + lhi * 8;
            const int o = sidx<AX>(m, col);
            dstR[o] = accR[v];
            dstI[o] = accI[v];
        }
    }
}

// ---------------------------------------------------------------------------
// Kernel 1: Taylor evolution. One block per time step. State in LDS (32 KB),
// accumulator in registers (16 complex elements / thread).
// ---------------------------------------------------------------------------
__global__ __launch_bounds__(256)
void tfim_evolve(const float* __restrict__ ts,
                 const float* __restrict__ px,
                 const float* __restrict__ pzz,
                 const int*   __restrict__ init_p,
                 float*       __restrict__ gstate /* [8][2][DIM] */) {
    __shared__ float tr[DIM], ti[DIM];

    const int   tid  = threadIdx.x;
    const float t    = ts[blockIdx.x];
    const int   init = init_p[0];

    float x[NQ];
#pragma unroll
    for (int i = 0; i < NQ; i++) x[i] = px[i];
    float zz[NQ - 1];
#pragma unroll
    for (int i = 0; i < NQ - 1; i++) zz[i] = pzz[i];

    float ar[16], ai[16], dg[16];
#pragma unroll
    for (int j = 0; j < 16; j++) {
        const int e = tid + j * 256;
        float d = 0.f;
#pragma unroll
        for (int i = 0; i < NQ - 1; i++) {
            const int b0 = (e >> i) & 1, b1 = (e >> (i + 1)) & 1;
            d += zz[i] * (float)((1 - 2 * b0) * (1 - 2 * b1));
        }
        dg[j] = d;
        const float v = (e == init) ? 1.f : 0.f;
        tr[e] = v; ti[e] = 0.f;
        ar[j] = v; ai[j] = 0.f;
    }
    __syncthreads();

    for (int k = 1; k <= TKAY; k++) {
        const float c = t / (float)k;
        float nr[16], ni[16];
#pragma unroll
        for (int j = 0; j < 16; j++) {
            const int e = tid + j * 256;
            float hr = dg[j] * tr[e];
            float hi = dg[j] * ti[e];
#pragma unroll
            for (int i = 0; i < NQ; i++) {
                const int f = e ^ (1 << i);
                hr += x[i] * tr[f];
                hi += x[i] * ti[f];
            }
            nr[j] = c * hi;       // (-i*c)*(hr + i*hi) = c*hi - i*c*hr
            ni[j] = -c * hr;
        }
        __syncthreads();
#pragma unroll
        for (int j = 0; j < 16; j++) {
            const int e = tid + j * 256;
            tr[e] = nr[j]; ti[e] = ni[j];
            ar[j] += nr[j]; ai[j] += ni[j];
        }
        __syncthreads();
    }

    float* outp = gstate + blockIdx.x * (2 * DIM);
#pragma unroll
    for (int j = 0; j < 16; j++) {
        const int e = tid + j * 256;
        outp[e]       = ar[j];
        outp[DIM + e] = ai[j];
    }
}

// ---------------------------------------------------------------------------
// Kernel 2: rotation via WMMA + probabilities + gather.
// One block per (t, obs) pair: 8 waves; 3 fully-unrolled axis applications
// (96 static v_wmma_f32_16x16x4_f32), ping-pong LDS buffers (64 KB).
// ---------------------------------------------------------------------------
__global__ __launch_bounds__(256)
void tfim_rotate_probs(const float* __restrict__ gstate,
                       const int*   __restrict__ pauli,   // (32, NQ)
                       const int*   __restrict__ indices, // (8, 32, 256)
                       float*       __restrict__ out) {   // (8, 32, 256)
    __shared__ float bufAr[DIM], bufAi[DIM], bufBr[DIM], bufBi[DIM];

    const int tid = threadIdx.x;
    const int tb  = blockIdx.x;       // t*32 + obs
    const int tt  = tb >> 5;
    const int ob  = tb & 31;

    // Load psi(t) into LDS (prefetch hint -> global_prefetch_b8).
    const float* st = gstate + tt * (2 * DIM);
    __builtin_prefetch(st, 0, 1);
    for (int e = tid; e < DIM; e += 256) {
        bufAr[e] = st[e];
        bufAi[e] = st[DIM + e];
    }
    __syncthreads();

    const int* pauli_row = pauli + ob * NQ;

    apply_axis<0>(bufAr, bufAi, bufBr, bufBi, pauli_row);
    __syncthreads();
    apply_axis<1>(bufBr, bufBi, bufAr, bufAi, pauli_row);
    __syncthreads();
    apply_axis<2>(bufAr, bufAi, bufBr, bufBi, pauli_row);
    __syncthreads();

    // Probabilities (final rotated state lives in bufB), then gather shots.
    for (int e = tid; e < DIM; e += 256) {
        const float r = bufBr[e], q = bufBi[e];
        bufAr[e] = r * r + q * q;
    }
    __syncthreads();

    const int shot = indices[tb * 256 + tid] & (DIM - 1);
    out[tb * 256 + tid] = bufAr[shot];
}

// ---------------------------------------------------------------------------
extern "C" void kernel_launch(void* const* d_in, const int* in_sizes, int n_in,
                              void* d_out, int out_size, void* d_ws, size_t ws_size,
                              hipStream_t stream) {
    (void)in_sizes; (void)n_in; (void)out_size; (void)ws_size;

    const int*   init_p  = (const int*)  d_in[0];   // initial_state (1)
    const float* ts      = (const float*)d_in[1];   // (8)
    const int*   pauli   = (const int*)  d_in[2];   // (32, 12)
    const int*   indices = (const int*)  d_in[3];   // (8, 32, 256)
    const float* px      = (const float*)d_in[4];   // (12)
    const float* pzz     = (const float*)d_in[5];   // (11)
    float*       out     = (float*)d_out;           // (8, 32, 256)
    float*       gstate  = (float*)d_ws;            // 8 * 2 * 4096 floats = 256 KB

    tfim_evolve<<<8, 256, 0, stream>>>(ts, px, pzz, init_p, gstate);
    tfim_rotate_probs<<<256, 256, 0, stream>>>(gstate, pauli, indices, out);
}